// VariationalGCNEncoder_66142496358859
// MI455X (gfx1250) — compile-verified
//
#include <hip/hip_runtime.h>

typedef __attribute__((ext_vector_type(2))) float v2f;
typedef __attribute__((ext_vector_type(8))) float v8f;

// ---------------------------------------------------------------- utilities

__global__ void fill_f32(float* __restrict__ p, float v, int n) {
  int i = blockIdx.x * blockDim.x + threadIdx.x;
  if (i < n) p[i] = v;
}

// deg[dst[e]] += 1 (self-loop "1" is pre-filled)
__global__ void deg_count(const long long* __restrict__ ei, float* __restrict__ deg, int E) {
  int i = blockIdx.x * blockDim.x + threadIdx.x;
  if (i < E) unsafeAtomicAdd(&deg[(size_t)ei[(size_t)E + i]], 1.0f);
}

__global__ void rsqrt_inplace(float* __restrict__ p, int n) {
  int i = blockIdx.x * blockDim.x + threadIdx.x;
  if (i < n) p[i] = rsqrtf(p[i]);
}

// one wave per edge: out[dst] += h[src] * dinv[src]*dinv[dst]
template <int C>
__global__ void edge_agg(const long long* __restrict__ ei, const float* __restrict__ h,
                         const float* __restrict__ dinv, float* __restrict__ out, int E) {
  int wave = (blockIdx.x * blockDim.x + threadIdx.x) >> 5;
  if (wave >= E) return;                       // wave-uniform exit
  int lane = threadIdx.x & 31;
  long long s = ei[wave];
  long long d = ei[(size_t)E + wave];
  float nrm = dinv[(size_t)s] * dinv[(size_t)d];
  const float* __restrict__ hs = h + (size_t)s * C;
  float* __restrict__ od = out + (size_t)d * C;
#pragma unroll
  for (int c = lane; c < C; c += 32) {
    unsafeAtomicAdd(&od[c], hs[c] * nrm);
  }
}

// ---------------------------------------------------------------- WMMA GEMM
// Craw[M,N] = op(A)[M,K] @ B[K,N]  (op = optional ReLU on A loads), f32.
// One wave per 16x16 tile, K stepped by 4 via V_WMMA_F32_16X16X4_F32.
// Fused epilogue additionally writes Cinit = acc*dinv[m]^2 + bias[n]
// (the GCN self-loop + bias initialization of the aggregation buffer),
// saving a full re-read pass over the GEMM output.
//
// A frag (16x4): lanes 0-15 -> M=lane, K={k,k+1}; lanes 16-31 -> M=lane-16, K={k+2,k+3}
// B frag (4x16): symmetric (lane selects N, lane-half selects K pair)
// C/D (16x16): VGPR v holds row M = v + 8*(lane>=16), N = lane&15.
template <bool RELU>
__global__ void gemm16x16_wmma(const float* __restrict__ A, const float* __restrict__ B,
                               float* __restrict__ Craw, float* __restrict__ Cinit,
                               const float* __restrict__ dinv, const float* __restrict__ bias,
                               int K, int N, int nTilesN, int nTiles) {
  int wave = (blockIdx.x * blockDim.x + threadIdx.x) >> 5;
  if (wave >= nTiles) return;                  // wave-uniform exit (EXEC stays all-1s)
  int lane = threadIdx.x & 31;
  int half = lane >> 4;                        // 0 or 1
  int l16 = lane & 15;
  int tm = wave / nTilesN;
  int tn = wave - tm * nTilesN;

  int rowA = tm * 16 + l16;
  int colB = tn * 16 + l16;
  const float* __restrict__ Arow = A + (size_t)rowA * K;

  v8f acc = {};
  for (int k = 0; k < K; k += 4) {
    int ka = k + half * 2;
    v2f a, b;
    a.x = Arow[ka];
    a.y = Arow[ka + 1];
    if (RELU) { a.x = fmaxf(a.x, 0.0f); a.y = fmaxf(a.y, 0.0f); }
    b.x = B[(size_t)ka * N + colB];
    b.y = B[(size_t)(ka + 1) * N + colB];
    acc = __builtin_amdgcn_wmma_f32_16x16x4_f32(
        /*neg_a=*/false, a, /*neg_b=*/false, b,
        /*c_mod=*/(short)0, acc, /*reuse_a=*/false, /*reuse_b=*/false);
  }

  int n = tn * 16 + l16;
  int mbase = tm * 16 + half * 8;
  float bn = bias[n];
#pragma unroll
  for (int v = 0; v < 8; ++v) {
    int m = mbase + v;
    float val = acc[v];
    Craw[(size_t)m * N + n] = val;
    float di = dinv[m];
    Cinit[(size_t)m * N + n] = val * di * di + bn;
  }
}

// ---------------------------------------------------------------- launcher

static inline int cdiv_i(int a, int b) { return (a + b - 1) / b; }
static inline size_t align256(size_t x) { return (x + 255) & ~(size_t)255; }

extern "C" void kernel_launch(void* const* d_in, const int* in_sizes, int n_in,
                              void* d_out, int out_size, void* d_ws, size_t ws_size,
                              hipStream_t stream) {
  const float* x        = (const float*)d_in[0];
  const long long* ei   = (const long long*)d_in[1];
  const float* W1       = (const float*)d_in[2];
  const float* b1       = (const float*)d_in[3];
  const float* Wmu      = (const float*)d_in[4];
  const float* bmu      = (const float*)d_in[5];
  const float* Wls      = (const float*)d_in[6];
  const float* bls      = (const float*)d_in[7];

  const int IN_CH = 128, HID = 64, OUT = 32;
  const int Nn = in_sizes[0] / IN_CH;   // 100000
  const int E  = in_sizes[1] / 2;       // 1600000

  // workspace layout
  char* ws = (char*)d_ws;
  size_t off = 0;
  float* dinv = (float*)(ws + off); off = align256(off + (size_t)Nn * 4);
  float* hx   = (float*)(ws + off); off = align256(off + (size_t)Nn * HID * 4);
  float* a1   = (float*)(ws + off); off = align256(off + (size_t)Nn * HID * 4);
  float* hmu  = (float*)(ws + off); off = align256(off + (size_t)Nn * OUT * 4);
  float* hls  = (float*)(ws + off); off = align256(off + (size_t)Nn * OUT * 4);

  float* mu = (float*)d_out;
  float* ls = mu + (size_t)Nn * OUT;

  const int TB = 256;  // 8 waves per block (wave32)

  // 1) degrees (self-loop included) -> dinv = rsqrt(deg)
  fill_f32<<<cdiv_i(Nn, TB), TB, 0, stream>>>(dinv, 1.0f, Nn);
  deg_count<<<cdiv_i(E, TB), TB, 0, stream>>>(ei, dinv, E);
  rsqrt_inplace<<<cdiv_i(Nn, TB), TB, 0, stream>>>(dinv, Nn);

  // 2) hx = x @ W1 (f32 WMMA); fused epilogue writes a1 = hx*dinv^2 + b1
  {
    int nTilesN = HID / 16;
    int nTiles = (Nn / 16) * nTilesN;
    gemm16x16_wmma<false><<<cdiv_i(nTiles * 32, TB), TB, 0, stream>>>(
        x, W1, hx, a1, dinv, b1, IN_CH, HID, nTilesN, nTiles);
  }

  // 3) layer-1 edge scatter-add into a1
  edge_agg<64><<<cdiv_i(E * 32, TB), TB, 0, stream>>>(ei, hx, dinv, a1, E);

  // 4) second-layer transforms: ReLU fused into A loads, self-loop+bias
  //    init fused into epilogue (written straight into d_out)
  {
    int nTilesN = OUT / 16;
    int nTiles = (Nn / 16) * nTilesN;
    gemm16x16_wmma<true><<<cdiv_i(nTiles * 32, TB), TB, 0, stream>>>(
        a1, Wmu, hmu, mu, dinv, bmu, HID, OUT, nTilesN, nTiles);
    gemm16x16_wmma<true><<<cdiv_i(nTiles * 32, TB), TB, 0, stream>>>(
        a1, Wls, hls, ls, dinv, bls, HID, OUT, nTilesN, nTiles);
  }

  // 5) mu / logstd edge scatter-add directly into d_out
  edge_agg<32><<<cdiv_i(E * 32, TB), TB, 0, stream>>>(ei, hmu, dinv, mu, E);
  edge_agg<32><<<cdiv_i(E * 32, TB), TB, 0, stream>>>(ei, hls, dinv, ls, E);
}